// Decoder_3650722202031
// MI455X (gfx1250) — compile-verified
//
#include <hip/hip_runtime.h>
#include <math.h>

// Problem constants (match reference)
#define HD 512            // hidden dim
#define BD 128            // batch
#define VD 8000           // vocab
#define TP 20             // eval points
#define BH (BD * HD)      // 65536 floats per state buffer

typedef __attribute__((ext_vector_type(2))) float v2f;
typedef __attribute__((ext_vector_type(8))) float v8f;

// ---------------------------------------------------------------------------
// f32 WMMA: D(16x16,f32) = A(16x4,f32) * B(4x16,f32) + C
// A frag: lane<16 -> {A[m][k],A[m][k+1]}, lane>=16 -> {A[m][k+2],A[m][k+3]},
//         m = lane%16.  B = W^T, so B[kk][n] = W[n][kk]: identical addressing
//         pattern on W rows.  C/D: vgpr r -> row r + 8*(lane>=16), col lane%16.
// ---------------------------------------------------------------------------
__device__ __forceinline__ v8f wmma_f32_k4(v2f a, v2f b, v8f c) {
  return __builtin_amdgcn_wmma_f32_16x16x4_f32(
      /*neg_a=*/false, a, /*neg_b=*/false, b,
      /*c_mod=*/(short)0, c, /*reuse_a=*/false, /*reuse_b=*/false);
}

// Generic wave-level GEMM core: computes a (16*MF) x (16*NF) tile of
// A(MxK, row-major) @ W^T  where W is (N x K) row-major.
// UNROLL bounds k-loop unrolling to contain register pressure on big tiles.
template <int MF, int NF, int UNROLL>
__device__ __forceinline__ void gemm_core(const float* __restrict__ A,
                                          const float* __restrict__ W,
                                          int K, int m0, int n0,
                                          v8f acc[MF][NF]) {
  const int lane = threadIdx.x & 31;
  const int r16  = lane & 15;
  const int ks   = (lane >> 4) << 1;  // 0 for lanes 0-15, 2 for lanes 16-31

  const float* ap[MF];
  const float* bp[NF];
#pragma unroll
  for (int i = 0; i < MF; ++i)
    ap[i] = A + (size_t)(m0 + 16 * i + r16) * K + ks;
#pragma unroll
  for (int j = 0; j < NF; ++j)
    bp[j] = W + (size_t)(n0 + 16 * j + r16) * K + ks;

#pragma unroll UNROLL
  for (int k = 0; k < K; k += 4) {
    v2f af[MF], bf[NF];
#pragma unroll
    for (int i = 0; i < MF; ++i) af[i] = *(const v2f*)(ap[i] + k);
#pragma unroll
    for (int j = 0; j < NF; ++j) bf[j] = *(const v2f*)(bp[j] + k);
#pragma unroll
    for (int i = 0; i < MF; ++i)
#pragma unroll
      for (int j = 0; j < NF; ++j)
        acc[i][j] = wmma_f32_k4(af[i], bf[j], acc[i][j]);
  }
}

// ---------------------------------------------------------------------------
// Init: X = hidden, TRAJ[0] = hidden
// ---------------------------------------------------------------------------
__global__ void init_state(const float* __restrict__ hidden,
                           float* __restrict__ X, float* __restrict__ TRAJ0) {
  const int i = blockIdx.x * blockDim.x + threadIdx.x;
  if (i < BH) {
    const float v = hidden[i];
    X[i]     = v;
    TRAJ0[i] = v;
  }
}

// ---------------------------------------------------------------------------
// Stage GEMM 1: Hout = tanh(Xin @ W1^T + b1)   (128x512) @ (512x512)
// Wave tile 16x32 -> 8 m-groups * 16 n-groups = 128 waves = 16 blocks * 8 waves
// Latency-critical (76 serial f-evals), so keep the per-wave WMMA chain short.
// ---------------------------------------------------------------------------
__global__ __launch_bounds__(256) void ode_gemm_tanh(
    const float* __restrict__ Xin, const float* __restrict__ W1,
    const float* __restrict__ b1, float* __restrict__ Hout) {
  const int wave = blockIdx.x * (blockDim.x >> 5) + (threadIdx.x >> 5);
  const int mg = wave >> 4;         // 0..7
  const int ng = wave & 15;         // 0..15
  const int m0 = mg * 16, n0 = ng * 32;

  const v8f z = {0.f, 0.f, 0.f, 0.f, 0.f, 0.f, 0.f, 0.f};
  v8f acc[1][2];
  acc[0][0] = z;
  acc[0][1] = z;
  gemm_core<1, 2, 4>(Xin, W1, HD, m0, n0, acc);

  const int lane  = threadIdx.x & 31;
  const int rbase = (lane >> 4) << 3;
  const int nlo   = lane & 15;
#pragma unroll
  for (int j = 0; j < 2; ++j) {
    const int col   = n0 + 16 * j + nlo;
    const float bia = b1[col];
#pragma unroll
    for (int r = 0; r < 8; ++r) {
      const int row = m0 + rbase + r;
      Hout[row * HD + col] = tanhf(acc[0][j][r] + bia);
    }
  }
}

// ---------------------------------------------------------------------------
// Stage GEMM 2 + fused RK4 bookkeeping (STAGE is a compile-time constant so
// the epilogue has no divergent/uniform branches):
//   k = Hin @ W2^T + b2
//   STAGE 0: ACC = k;        XIN = X + 0.5*dt*k
//   STAGE 1: ACC += 2k;      XIN = X + 0.5*dt*k
//   STAGE 2: ACC += 2k;      XIN = X + dt*k
//   STAGE 3: Xn = X + dt/6*(ACC + k); X = Xn; TRAJ_slot = Xn
// ---------------------------------------------------------------------------
template <int STAGE>
__global__ __launch_bounds__(256) void ode_gemm_stage(
    const float* __restrict__ Hin, const float* __restrict__ W2,
    const float* __restrict__ b2, float* __restrict__ X,
    float* __restrict__ ACC, float* __restrict__ XIN,
    float* __restrict__ TRAJ_slot, const float* __restrict__ t_span,
    int step) {
  const int wave = blockIdx.x * (blockDim.x >> 5) + (threadIdx.x >> 5);
  const int mg = wave >> 4;
  const int ng = wave & 15;
  const int m0 = mg * 16, n0 = ng * 32;

  const v8f z = {0.f, 0.f, 0.f, 0.f, 0.f, 0.f, 0.f, 0.f};
  v8f acc[1][2];
  acc[0][0] = z;
  acc[0][1] = z;
  gemm_core<1, 2, 4>(Hin, W2, HD, m0, n0, acc);

  const float dt = t_span[step + 1] - t_span[step];

  const int lane  = threadIdx.x & 31;
  const int rbase = (lane >> 4) << 3;
  const int nlo   = lane & 15;
#pragma unroll
  for (int j = 0; j < 2; ++j) {
    const int col   = n0 + 16 * j + nlo;
    const float bia = b2[col];
#pragma unroll
    for (int r = 0; r < 8; ++r) {
      const int row = m0 + rbase + r;
      const int idx = row * HD + col;
      const float kv = acc[0][j][r] + bia;
      const float xv = X[idx];
      if (STAGE == 0) {
        ACC[idx] = kv;
        XIN[idx] = xv + 0.5f * dt * kv;
      } else if (STAGE == 1) {
        ACC[idx] += 2.0f * kv;
        XIN[idx] = xv + 0.5f * dt * kv;
      } else if (STAGE == 2) {
        ACC[idx] += 2.0f * kv;
        XIN[idx] = xv + dt * kv;
      } else {
        const float xn = xv + (dt * (1.0f / 6.0f)) * (ACC[idx] + kv);
        X[idx]         = xn;
        TRAJ_slot[idx] = xn;
      }
    }
  }
}

// ---------------------------------------------------------------------------
// Final projection: Out(2560 x 8000) = TRAJ(2560 x 512) @ Wfc^T + bfc
// Wave tile 64x64 (4x4 frags): 8 loads feed 16 WMMAs per k-step, and total
// L2 read traffic drops to ~1.3 GB (A 5.24MB*125 + Wfc 16.4MB*40) vs 2.6 GB
// at 32x32.  40 * 125 = 5000 waves = 625 blocks * 8 waves.
// Non-temporal stores keep the 82 MB output stream from evicting Wfc
// (16 MB, resident in the 192 MB L2).
// ---------------------------------------------------------------------------
__global__ __launch_bounds__(256) void proj_gemm(
    const float* __restrict__ TRAJ, const float* __restrict__ Wfc,
    const float* __restrict__ bfc, float* __restrict__ Out) {
  const int wave = blockIdx.x * (blockDim.x >> 5) + (threadIdx.x >> 5);
  const int NG = VD / 64;  // 125
  const int mg = wave / NG;
  const int ng = wave % NG;
  const int m0 = mg * 64, n0 = ng * 64;

  const v8f z = {0.f, 0.f, 0.f, 0.f, 0.f, 0.f, 0.f, 0.f};
  v8f acc[4][4];
#pragma unroll
  for (int i = 0; i < 4; ++i)
#pragma unroll
    for (int j = 0; j < 4; ++j) acc[i][j] = z;

  gemm_core<4, 4, 2>(TRAJ, Wfc, HD, m0, n0, acc);

  const int lane  = threadIdx.x & 31;
  const int rbase = (lane >> 4) << 3;
  const int nlo   = lane & 15;
#pragma unroll
  for (int j = 0; j < 4; ++j) {
    const int col   = n0 + 16 * j + nlo;
    const float bia = bfc[col];
#pragma unroll
    for (int i = 0; i < 4; ++i) {
#pragma unroll
      for (int r = 0; r < 8; ++r) {
        const int row = m0 + 16 * i + rbase + r;
        __builtin_nontemporal_store(acc[i][j][r] + bia,
                                    &Out[(size_t)row * VD + col]);
      }
    }
  }
}

// ---------------------------------------------------------------------------
// Host launcher
// Inputs: 0 hidden(1,128,512) 1 t_span(20) 2 W1(512,512) 3 b1(512)
//         4 W2(512,512) 5 b2(512) 6 Wfc(8000,512) 7 bfc(8000)
// Output: (20,128,8000) f32
// ---------------------------------------------------------------------------
extern "C" void kernel_launch(void* const* d_in, const int* in_sizes, int n_in,
                              void* d_out, int out_size, void* d_ws,
                              size_t ws_size, hipStream_t stream) {
  (void)in_sizes; (void)n_in; (void)out_size; (void)ws_size;

  const float* hidden = (const float*)d_in[0];
  const float* t_span = (const float*)d_in[1];
  const float* W1     = (const float*)d_in[2];
  const float* b1     = (const float*)d_in[3];
  const float* W2     = (const float*)d_in[4];
  const float* b2     = (const float*)d_in[5];
  const float* Wfc    = (const float*)d_in[6];
  const float* bfc    = (const float*)d_in[7];
  float*       out    = (float*)d_out;

  // Workspace layout (floats): X | XIN | H | ACC | TRAJ(20*BH)  ~= 6.3 MB
  float* ws   = (float*)d_ws;
  float* X    = ws;
  float* XIN  = ws + (size_t)BH;
  float* Hbuf = ws + (size_t)2 * BH;
  float* ACC  = ws + (size_t)3 * BH;
  float* TRAJ = ws + (size_t)4 * BH;

  init_state<<<BH / 256, 256, 0, stream>>>(hidden, X, TRAJ);

  for (int step = 0; step < TP - 1; ++step) {
    float* traj_slot = TRAJ + (size_t)(step + 1) * BH;

    // stage 0 (reads X)
    ode_gemm_tanh<<<16, 256, 0, stream>>>(X, W1, b1, Hbuf);
    ode_gemm_stage<0><<<16, 256, 0, stream>>>(Hbuf, W2, b2, X, ACC, XIN,
                                              traj_slot, t_span, step);
    // stage 1 (reads XIN)
    ode_gemm_tanh<<<16, 256, 0, stream>>>(XIN, W1, b1, Hbuf);
    ode_gemm_stage<1><<<16, 256, 0, stream>>>(Hbuf, W2, b2, X, ACC, XIN,
                                              traj_slot, t_span, step);
    // stage 2
    ode_gemm_tanh<<<16, 256, 0, stream>>>(XIN, W1, b1, Hbuf);
    ode_gemm_stage<2><<<16, 256, 0, stream>>>(Hbuf, W2, b2, X, ACC, XIN,
                                              traj_slot, t_span, step);
    // stage 3 (writes X and TRAJ[step+1])
    ode_gemm_tanh<<<16, 256, 0, stream>>>(XIN, W1, b1, Hbuf);
    ode_gemm_stage<3><<<16, 256, 0, stream>>>(Hbuf, W2, b2, X, ACC, XIN,
                                              traj_slot, t_span, step);
  }

  // 2560 x 8000 projection: 40 m-groups * 125 n-groups / 8 waves per block
  proj_gemm<<<625, 256, 0, stream>>>(TRAJ, Wfc, bfc, out);
}